// GlobalAttentionTransformerEncoder_57741540327437
// MI455X (gfx1250) — compile-verified
//
#include <hip/hip_runtime.h>
#include <math.h>

// ---------------------------------------------------------------------------
// GlobalAttentionTransformerEncoder for MI455X (gfx1250, wave32, WMMA)
// B=2 S=1024 E=512 H=8 G=32 head_dim=64
// All heavy math as V_WMMA_F32_16X16X4_F32 with causal masking; the only
// serial piece (cumsum over S) runs as 512 independent lane chains.
// ---------------------------------------------------------------------------

typedef __attribute__((ext_vector_type(2))) float v2f;
typedef __attribute__((ext_vector_type(8))) float v8f;

#define Bb 2
#define Ss 1024
#define Ee 512
#define Hh 8
#define Gg 32
#define Dd 64
#define MIN_NORM 1e-8f

__device__ __forceinline__ v8f zero8() {
  v8f z;
#pragma unroll
  for (int i = 0; i < 8; ++i) z[i] = 0.f;
  return z;
}

// D = A(16x4) * B(4x16) + C, fp32 WMMA (wave32).
// A frag: lane L holds M=L&15, K = (L>>4)*2 + {0,1}. B frag mirrored with N.
__device__ __forceinline__ v8f wmma4(v2f a, v2f b, v8f c) {
  return __builtin_amdgcn_wmma_f32_16x16x4_f32(
      /*neg_a=*/false, a, /*neg_b=*/false, b,
      /*c_mod=*/(short)0, c, /*reuse_a=*/false, /*reuse_b=*/false);
}

// ---------------------------------------------------------------------------
// K0: zero the global mean accumulator
// ---------------------------------------------------------------------------
__global__ void init_ws(float* wsum) {
  if (threadIdx.x == 0) wsum[0] = 0.f;
}

// ---------------------------------------------------------------------------
// K1/K5: generic [M,512] @ [512,512] GEMM, wave computes a 16x64 C tile.
// MODE 0: X is [B*S,512] rows; Y scattered to [B,H,S,64] (+[B,H,64,S] if WT)
// MODE 1: X is static_context [G,512], rows broadcast over B; Y -> [B,H,G,64]
// MODE 2: X gathered from out_heads [B,H,S,64] (concat heads); Y -> [B,S,512]
// Templated so the hot K-loop is branch-free straight-line code.
// ---------------------------------------------------------------------------
template <int MODE, bool WT>
__global__ void gemm512(const float* __restrict__ X, const float* __restrict__ W,
                        float* __restrict__ Y, float* __restrict__ YT, int M) {
  const int lane = threadIdx.x & 31;
  const int wid = blockIdx.x * (blockDim.x >> 5) + (threadIdx.x >> 5);
  const int tiles_m = M >> 4;
  const int tm = wid >> 3;  // 8 column groups of 64
  const int tn = wid & 7;
  if (tm >= tiles_m) return;

  const int ml = lane & 15;
  const int hi = lane >> 4;
  const int kb = hi * 2;
  const int m = tm * 16 + ml;

  v8f acc[4];
#pragma unroll
  for (int j = 0; j < 4; ++j) acc[j] = zero8();

  for (int k0 = 0; k0 < Ee; k0 += 4) {
    const int ka = k0 + kb;
    float a0, a1;
    if (MODE == 0) {
      a0 = X[(size_t)m * Ee + ka];
      a1 = X[(size_t)m * Ee + ka + 1];
    } else if (MODE == 1) {
      const int g = m & (Gg - 1);
      a0 = X[(size_t)g * Ee + ka];
      a1 = X[(size_t)g * Ee + ka + 1];
    } else {
      const int b = m >> 10, s = m & (Ss - 1);
      const int h0 = ka >> 6, e0 = ka & 63;
      a0 = X[(((size_t)(b * Hh + h0)) * Ss + s) * Dd + e0];
      const int h1 = (ka + 1) >> 6, e1 = (ka + 1) & 63;
      a1 = X[(((size_t)(b * Hh + h1)) * Ss + s) * Dd + e1];
    }
    v2f a;
    a.x = a0;
    a.y = a1;
    const float* wp0 = W + (size_t)ka * Ee + tn * 64 + ml;
    const float* wp1 = wp0 + Ee;
#pragma unroll
    for (int j = 0; j < 4; ++j) {
      v2f b;
      b.x = wp0[j * 16];
      b.y = wp1[j * 16];
      acc[j] = wmma4(a, b, acc[j]);
    }
  }

#pragma unroll
  for (int j = 0; j < 4; ++j) {
#pragma unroll
    for (int r = 0; r < 8; ++r) {
      const int mm = tm * 16 + r + 8 * hi;
      const int n = tn * 64 + j * 16 + ml;
      const float val = acc[j][r];
      if (MODE == 0) {
        const int b = mm >> 10, s = mm & (Ss - 1);
        const int h = n >> 6, e = n & 63;
        Y[(((size_t)(b * Hh + h)) * Ss + s) * Dd + e] = val;
        if (WT) YT[(((size_t)(b * Hh + h)) * Dd + e) * Ss + s] = val;
      } else if (MODE == 1) {
        const int b = mm >> 5, g = mm & (Gg - 1);
        const int h = n >> 6, e = n & 63;
        Y[(((size_t)(b * Hh + h)) * Gg + g) * Dd + e] = val;
      } else {
        Y[(size_t)mm * Ee + n] = val;
      }
    }
  }
}

// ---------------------------------------------------------------------------
// K2: w_raw[b,h,s,g] = (ks[b,h,s,:] . cs[b,h,g,:]) / 8   (+ global sum for mean)
// Wave computes a 16(s) x 32(g) tile; K = 64.
// ---------------------------------------------------------------------------
__global__ void wraw_kernel(const float* __restrict__ ks,
                            const float* __restrict__ cs,
                            float* __restrict__ wraw, float* wsum) {
  const int lane = threadIdx.x & 31;
  const int wid = blockIdx.x * (blockDim.x >> 5) + (threadIdx.x >> 5);
  const int bh = wid >> 6;
  const int s0 = (wid & 63) << 4;
  const int ml = lane & 15, hi = lane >> 4, kb = hi * 2;

  const float* kbase = ks + (size_t)bh * Ss * Dd;
  const float* cbase = cs + (size_t)bh * Gg * Dd;

  v8f acc[2];
  acc[0] = zero8();
  acc[1] = zero8();

  for (int k0 = 0; k0 < Dd; k0 += 4) {
    const int ka = k0 + kb;
    v2f a;
    a.x = kbase[(size_t)(s0 + ml) * Dd + ka];
    a.y = kbase[(size_t)(s0 + ml) * Dd + ka + 1];
#pragma unroll
    for (int j = 0; j < 2; ++j) {
      v2f b;
      b.x = cbase[(size_t)(j * 16 + ml) * Dd + ka];
      b.y = cbase[(size_t)(j * 16 + ml) * Dd + ka + 1];
      acc[j] = wmma4(a, b, acc[j]);
    }
  }

  float part = 0.f;
#pragma unroll
  for (int j = 0; j < 2; ++j) {
#pragma unroll
    for (int r = 0; r < 8; ++r) {
      const int s = s0 + r + 8 * hi;
      const int g = j * 16 + ml;
      const float val = acc[j][r] * 0.125f;  // 1/sqrt(64)
      wraw[((size_t)bh * Ss + s) * Gg + g] = val;
      part += val;
    }
  }
  atomicAdd(wsum, part);
}

// ---------------------------------------------------------------------------
// K3: w = exp(w_raw - mean); norms = clamp(cumsum_s(w)); also wT transposed.
// One wave per (b,h), lane = g. Chunked: clause-load 32 s-steps into registers,
// run the serial cumsum chain, and emit wT via a 32x33 LDS transpose so the
// transposed stores are 128B-coalesced instead of 4KB-strided scatters.
// ---------------------------------------------------------------------------
__global__ void scan_kernel(const float* __restrict__ wraw,
                            const float* __restrict__ wsum,
                            float* __restrict__ wexp, float* __restrict__ wT,
                            float* __restrict__ nrm) {
  __shared__ float tile[32][33];
  const int bh = blockIdx.x;
  const int lane = threadIdx.x;  // 0..31 == g
  const float mean = wsum[0] * (1.0f / ((float)Bb * Hh * Ss * Gg));
  float run = 0.f;
  const size_t base = (size_t)bh * Ss * Gg;
  for (int c = 0; c < Ss; c += 32) {
    float vals[32];
#pragma unroll
    for (int i = 0; i < 32; ++i)
      vals[i] = wraw[base + (size_t)(c + i) * Gg + lane];
#pragma unroll
    for (int i = 0; i < 32; ++i) {
      const float v = expf(vals[i] - mean);
      run += v;
      const float nm = (run <= MIN_NORM) ? MIN_NORM : run;
      wexp[base + (size_t)(c + i) * Gg + lane] = v;
      nrm[base + (size_t)(c + i) * Gg + lane] = nm;
      tile[i][lane] = v;
    }
    __builtin_amdgcn_wave_barrier();
#pragma unroll
    for (int gg = 0; gg < 32; ++gg)  // coalesced: 32 lanes cover 32 s values
      wT[((size_t)bh * Gg + gg) * Ss + c + lane] = tile[lane][gg];
    __builtin_amdgcn_wave_barrier();
  }
}

// ---------------------------------------------------------------------------
// K4: fused causal attention per 16-row s-tile (one wave per tile).
// Phase A: L[s,g] = sum_{t<=s} (q_s.k_t) * w[t,g] ; logits = L / norms
//          aw' = softmax_g(logits) / norms
// Phase B: out[s,e] = sum_{t<=s} (aw' . wT[:,t]) * v[t,e]
// ---------------------------------------------------------------------------
__global__ void attn_kernel(const float* __restrict__ qs,
                            const float* __restrict__ kT,
                            const float* __restrict__ vs,
                            const float* __restrict__ wexp,
                            const float* __restrict__ wT,
                            const float* __restrict__ nrm,
                            float* __restrict__ oh) {
  __shared__ float lt[8][16][17];  // per-wave 16x16 tile pass-through
  __shared__ float lg[8][16][33];  // per-wave logits / aw'

  const int lane = threadIdx.x & 31;
  const int wv = threadIdx.x >> 5;
  const int wid = blockIdx.x * 8 + wv;
  const int bh = wid >> 6;
  const int s0 = (wid & 63) << 4;
  const int ml = lane & 15, hi = lane >> 4, kb = hi * 2;

  const float* qsb = qs + (size_t)bh * Ss * Dd;
  const float* kTb = kT + (size_t)bh * Dd * Ss;
  const float* vsb = vs + (size_t)bh * Ss * Dd;
  const float* wb = wexp + (size_t)bh * Ss * Gg;
  const float* wTb = wT + (size_t)bh * Gg * Ss;
  const float* nb = nrm + (size_t)bh * Ss * Gg;

  // ---- Phase A: logits ----
  v2f qf[16];
#pragma unroll
  for (int kk = 0; kk < 16; ++kk) {
    const int ka = kk * 4 + kb;
    qf[kk].x = qsb[(size_t)(s0 + ml) * Dd + ka];
    qf[kk].y = qsb[(size_t)(s0 + ml) * Dd + ka + 1];
  }

  v8f L0 = zero8(), L1 = zero8();
  for (int t0 = 0; t0 <= s0; t0 += 16) {
    if (t0 + 16 <= s0) {  // prefetch next t-tile streams into WGP cache
      __builtin_prefetch(&kTb[(size_t)kb * Ss + t0 + 16 + ml], 0, 1);
      __builtin_prefetch(&wb[(size_t)(t0 + 16) * Gg + lane], 0, 1);
    }
    v8f A = zero8();
#pragma unroll
    for (int kk = 0; kk < 16; ++kk) {  // K = 64: A = Q_tile . K_tile^T
      const int ka = kk * 4 + kb;
      v2f b;
      b.x = kTb[(size_t)ka * Ss + t0 + ml];
      b.y = kTb[(size_t)(ka + 1) * Ss + t0 + ml];
      A = wmma4(qf[kk], b, A);
    }
    if (t0 == s0) {  // causal mask on diagonal tile (keep t<=s, i.e. n<=m)
#pragma unroll
      for (int r = 0; r < 8; ++r)
        if (ml > r + 8 * hi) A[r] = 0.f;
    }
#pragma unroll
    for (int r = 0; r < 8; ++r) lt[wv][r + 8 * hi][ml] = A[r];
    __builtin_amdgcn_wave_barrier();
#pragma unroll
    for (int kk = 0; kk < 4; ++kk) {  // K = 16: L += A_tile . w[t0:t0+16, :]
      const int ka = kk * 4 + kb;
      v2f a2;
      a2.x = lt[wv][ml][ka];
      a2.y = lt[wv][ml][ka + 1];
      v2f b0, b1;
      b0.x = wb[(size_t)(t0 + ka) * Gg + ml];
      b0.y = wb[(size_t)(t0 + ka + 1) * Gg + ml];
      b1.x = wb[(size_t)(t0 + ka) * Gg + 16 + ml];
      b1.y = wb[(size_t)(t0 + ka + 1) * Gg + 16 + ml];
      L0 = wmma4(a2, b0, L0);
      L1 = wmma4(a2, b1, L1);
    }
    __builtin_amdgcn_wave_barrier();
  }

  // logits = L / norms -> LDS
#pragma unroll
  for (int r = 0; r < 8; ++r) {
    const int mm = r + 8 * hi;
    const int srow = s0 + mm;
    lg[wv][mm][ml] = L0[r] / nb[(size_t)srow * Gg + ml];
    lg[wv][mm][16 + ml] = L1[r] / nb[(size_t)srow * Gg + 16 + ml];
  }
  __builtin_amdgcn_wave_barrier();

  // softmax over g (lanes 0..15 each own one row), then aw' = aw / norms
  if (hi == 0) {
    const int mm = ml;
    const int srow = s0 + mm;
    float mx = -3.0e38f;
    for (int g = 0; g < Gg; ++g) mx = fmaxf(mx, lg[wv][mm][g]);
    float ssum = 0.f;
    for (int g = 0; g < Gg; ++g) {
      const float e = expf(lg[wv][mm][g] - mx);
      ssum += e;
      lg[wv][mm][g] = e;
    }
    const float inv = 1.f / ssum;
    for (int g = 0; g < Gg; ++g)
      lg[wv][mm][g] = lg[wv][mm][g] * inv / nb[(size_t)srow * Gg + g];
  }
  __builtin_amdgcn_wave_barrier();

  // ---- Phase B: output ----
  v2f af[8];
#pragma unroll
  for (int kk = 0; kk < 8; ++kk) {
    const int ka = kk * 4 + kb;
    af[kk].x = lg[wv][ml][ka];
    af[kk].y = lg[wv][ml][ka + 1];
  }

  v8f O[4];
#pragma unroll
  for (int j = 0; j < 4; ++j) O[j] = zero8();

  for (int t0 = 0; t0 <= s0; t0 += 16) {
    if (t0 + 16 <= s0) {  // prefetch next t-tile streams
      __builtin_prefetch(&wTb[(size_t)kb * Ss + t0 + 16 + ml], 0, 1);
      __builtin_prefetch(&vsb[(size_t)(t0 + 16) * Dd + lane], 0, 1);
    }
    v8f P = zero8();
#pragma unroll
    for (int kk = 0; kk < 8; ++kk) {  // K = 32: P = aw' . w^T[:, t0:t0+16]
      const int ka = kk * 4 + kb;
      v2f b;
      b.x = wTb[(size_t)ka * Ss + t0 + ml];
      b.y = wTb[(size_t)(ka + 1) * Ss + t0 + ml];
      P = wmma4(af[kk], b, P);
    }
    if (t0 == s0) {
#pragma unroll
      for (int r = 0; r < 8; ++r)
        if (ml > r + 8 * hi) P[r] = 0.f;
    }
#pragma unroll
    for (int r = 0; r < 8; ++r) lt[wv][r + 8 * hi][ml] = P[r];
    __builtin_amdgcn_wave_barrier();
#pragma unroll
    for (int kk = 0; kk < 4; ++kk) {  // K = 16: O += P_tile . V_tile(16x64)
      const int ka = kk * 4 + kb;
      v2f a2;
      a2.x = lt[wv][ml][ka];
      a2.y = lt[wv][ml][ka + 1];
      const float* vr0 = vsb + (size_t)(t0 + ka) * Dd;
      const float* vr1 = vsb + (size_t)(t0 + ka + 1) * Dd;
#pragma unroll
      for (int j = 0; j < 4; ++j) {
        v2f b;
        b.x = vr0[j * 16 + ml];
        b.y = vr1[j * 16 + ml];
        O[j] = wmma4(a2, b, O[j]);
      }
    }
    __builtin_amdgcn_wave_barrier();
  }

  float* ohb = oh + (size_t)bh * Ss * Dd;
#pragma unroll
  for (int j = 0; j < 4; ++j) {
#pragma unroll
    for (int r = 0; r < 8; ++r) {
      const int mm = r + 8 * hi;
      ohb[(size_t)(s0 + mm) * Dd + j * 16 + ml] = O[j][r];
    }
  }
}

// ---------------------------------------------------------------------------
// Launch pipeline
// ---------------------------------------------------------------------------
extern "C" void kernel_launch(void* const* d_in, const int* in_sizes, int n_in,
                              void* d_out, int out_size, void* d_ws,
                              size_t ws_size, hipStream_t stream) {
  const float* query = (const float*)d_in[0];
  const float* key = (const float*)d_in[1];
  const float* value = (const float*)d_in[2];
  const float* sctx = (const float*)d_in[3];
  const float* qproj = (const float*)d_in[4];
  const float* kproj = (const float*)d_in[5];
  const float* vproj = (const float*)d_in[6];
  const float* oproj = (const float*)d_in[7];
  const float* cproj = (const float*)d_in[8];
  float* out = (float*)d_out;
  float* ws = (float*)d_ws;

  const size_t BHSD = (size_t)Bb * Hh * Ss * Dd;  // 1,048,576
  const size_t BHSG = (size_t)Bb * Hh * Ss * Gg;  // 524,288
  float* qs = ws;
  float* ks = qs + BHSD;
  float* vs = ks + BHSD;
  float* kT = vs + BHSD;
  float* cs = kT + BHSD;
  float* wraw = cs + (size_t)Bb * Hh * Gg * Dd;
  float* wexp = wraw + BHSG;
  float* wT = wexp + BHSG;
  float* nrm = wT + BHSG;
  float* oh = nrm + BHSG;
  float* wsum = oh + BHSD;

  init_ws<<<1, 32, 0, stream>>>(wsum);
  // head projections: [2048,512]@[512,512] each (WMMA f32), k also transposed
  gemm512<0, false><<<128, 256, 0, stream>>>(query, qproj, qs, nullptr, Bb * Ss);
  gemm512<0, true><<<128, 256, 0, stream>>>(key, kproj, ks, kT, Bb * Ss);
  gemm512<0, false><<<128, 256, 0, stream>>>(value, vproj, vs, nullptr, Bb * Ss);
  gemm512<1, false><<<4, 256, 0, stream>>>(sctx, cproj, cs, nullptr, Bb * Gg);
  // w_raw = cs.ks / sqrt(64), plus global-mean accumulation
  wraw_kernel<<<128, 256, 0, stream>>>(ks, cs, wraw, wsum);
  // exp + cumsum norms (+ transposed w, coalesced via LDS transpose)
  scan_kernel<<<Bb * Hh, 32, 0, stream>>>(wraw, wsum, wexp, wT, nrm);
  // fused causal attention core
  attn_kernel<<<128, 256, 0, stream>>>(qs, kT, vs, wexp, wT, nrm, oh);
  // output projection -> d_out [B,S,E]
  gemm512<2, false><<<128, 256, 0, stream>>>(oh, oproj, out, nullptr, Bb * Ss);
}